// YoloLoss_43190191128633
// MI455X (gfx1250) — compile-verified
//
#include <hip/hip_runtime.h>

#define LAMBDA_NOOBJ 0.5f
#define LAMBDA_COORD 5.0f

typedef __attribute__((ext_vector_type(2))) float v2f;
typedef __attribute__((ext_vector_type(4))) float f4;
typedef __attribute__((ext_vector_type(8))) float v8f;

// Exact f32 wave32 sum using one V_WMMA_F32_16X16X4_F32:
//   A = (acc, 0) per lane  -> A is a 16x4 f32 matrix holding every lane's acc
//   B = all ones (4x16)    -> D[m,n] = rowsum_m(A), independent of n
// Lane l<16 holds D rows 0-7 of column l in its 8 regs; lanes 16-31 hold rows
// 8-15. Per-lane sum of 8 regs + shfl_xor(16) == sum of all A entries == wave
// sum, in every lane. Requires EXEC == all-ones (guaranteed by uniform CF).
__device__ __forceinline__ float wave_reduce_f32(float v) {
  v2f a; a[0] = v;    a[1] = 0.0f;
  v2f b; b[0] = 1.0f; b[1] = 1.0f;
  v8f c = {};
  c = __builtin_amdgcn_wmma_f32_16x16x4_f32(
      /*neg_a=*/false, a, /*neg_b=*/false, b,
      /*c_mod=*/(short)0, c, /*reuse_a=*/false, /*reuse_b=*/false);
  float s = ((c[0] + c[1]) + (c[2] + c[3])) + ((c[4] + c[5]) + (c[6] + c[7]));
  s += __shfl_xor(s, 16, 32);
  return s;
}

// Stage 1: stream 262 MB with NT B128 loads, accumulate 3 masked sums,
// reduce per-wave (WMMA) then per-block (LDS), emit 3 partials per block.
__global__ void __launch_bounds__(256)
yolo_partial_kernel(const float* __restrict__ pred,
                    const float* __restrict__ targ,
                    float* __restrict__ partial,   // [gridDim.x * 3]
                    long long ngroups)             // groups of 4 cells (20 floats)
{
  const f4* __restrict__ pv = reinterpret_cast<const f4*>(pred);
  const f4* __restrict__ tv = reinterpret_cast<const f4*>(targ);

  float accBox = 0.0f, accObjC = 0.0f, accNoObjC = 0.0f;

  const long long stride = (long long)gridDim.x * (long long)blockDim.x;
  long long g = (long long)blockIdx.x * blockDim.x + threadIdx.x;

  for (; g < ngroups; g += stride) {
    // Prefetch next grid-stride tile (global_prefetch_b8).
    long long gn = g + stride;
    if (gn < ngroups) {
      __builtin_prefetch((const void*)(pv + gn * 5), 0, 0);
      __builtin_prefetch((const void*)(tv + gn * 5), 0, 0);
    }

    // 4 cells * 5 channels = 20 floats = 5 aligned B128 loads per array.
    float pf[20], tf[20];
#pragma unroll
    for (int j = 0; j < 5; ++j) {
      f4 p = __builtin_nontemporal_load(pv + g * 5 + j);
      f4 t = __builtin_nontemporal_load(tv + g * 5 + j);
#pragma unroll
      for (int k = 0; k < 4; ++k) { pf[4 * j + k] = p[k]; tf[4 * j + k] = t[k]; }
    }

#pragma unroll
    for (int c = 0; c < 4; ++c) {
      const int b5 = 5 * c;
      const float dc = tf[b5] - pf[b5];
      const float cd = dc * dc;
      float bd = 0.0f;
#pragma unroll
      for (int k = 1; k < 5; ++k) {
        const float d = tf[b5 + k] - pf[b5 + k];
        bd = fmaf(d, d, bd);
      }
      const bool obj = (tf[b5] == 1.0f);
      accBox    += obj ? bd : 0.0f;
      accObjC   += obj ? cd : 0.0f;
      accNoObjC += obj ? 0.0f : cd;
    }
  }

  // Wave-level reduction via WMMA (EXEC all-ones: full 256-thread blocks,
  // reconverged after the loop).
  const float wBox = wave_reduce_f32(accBox);
  const float wObj = wave_reduce_f32(accObjC);
  const float wNo  = wave_reduce_f32(accNoObjC);

  __shared__ float smem[8 * 3];  // 8 waves per 256-thread block
  const int lane = threadIdx.x & 31;
  const int wave = threadIdx.x >> 5;
  if (lane == 0) {
    smem[wave * 3 + 0] = wBox;
    smem[wave * 3 + 1] = wObj;
    smem[wave * 3 + 2] = wNo;
  }
  __syncthreads();
  if (threadIdx.x == 0) {
    float b = 0.0f, o = 0.0f, n = 0.0f;
#pragma unroll
    for (int w = 0; w < 8; ++w) {
      b += smem[w * 3 + 0];
      o += smem[w * 3 + 1];
      n += smem[w * 3 + 2];
    }
    partial[(long long)blockIdx.x * 3 + 0] = b;
    partial[(long long)blockIdx.x * 3 + 1] = o;
    partial[(long long)blockIdx.x * 3 + 2] = n;
  }
}

// Stage 2: single block folds the per-block partials and writes the 4 outputs.
__global__ void __launch_bounds__(256)
yolo_final_kernel(const float* __restrict__ partial, int nparts,
                  float* __restrict__ out)
{
  float b = 0.0f, o = 0.0f, n = 0.0f;
  // nparts (1280) % blockDim (256) == 0 -> uniform trip count, EXEC all-ones.
  for (int i = threadIdx.x; i < nparts; i += blockDim.x) {
    b += partial[i * 3 + 0];
    o += partial[i * 3 + 1];
    n += partial[i * 3 + 2];
  }
  b = wave_reduce_f32(b);
  o = wave_reduce_f32(o);
  n = wave_reduce_f32(n);

  __shared__ float smem[8 * 3];
  const int lane = threadIdx.x & 31;
  const int wave = threadIdx.x >> 5;
  if (lane == 0) {
    smem[wave * 3 + 0] = b;
    smem[wave * 3 + 1] = o;
    smem[wave * 3 + 2] = n;
  }
  __syncthreads();
  if (threadIdx.x == 0) {
    float B = 0.0f, O = 0.0f, N = 0.0f;
#pragma unroll
    for (int w = 0; w < 8; ++w) {
      B += smem[w * 3 + 0];
      O += smem[w * 3 + 1];
      N += smem[w * 3 + 2];
    }
    const float box_loss       = LAMBDA_COORD * B;
    const float object_loss    = O;
    const float no_object_loss = LAMBDA_NOOBJ * N;
    out[0] = box_loss + object_loss + no_object_loss;
    out[1] = box_loss;
    out[2] = object_loss;
    out[3] = no_object_loss;
  }
}

extern "C" void kernel_launch(void* const* d_in, const int* in_sizes, int n_in,
                              void* d_out, int out_size, void* d_ws, size_t ws_size,
                              hipStream_t stream) {
  const float* pred = (const float*)d_in[0];   // [65536,10,10,5] f32
  const float* targ = (const float*)d_in[1];   // [65536,10,10,5] f32
  float* out = (float*)d_out;                  // 4 floats
  float* partial = (float*)d_ws;               // blocks*3 floats scratch

  // 32,768,000 floats / 20 floats-per-group = 1,638,400 four-cell groups.
  const long long ngroups = (long long)in_sizes[0] / 20;

  // 1280 * 256 = 327,680 threads; 1,638,400 / 327,680 = 5 iterations each:
  // perfectly uniform trip counts (no divergence -> WMMA EXEC requirement ok).
  const int blocks = 1280;
  const int threads = 256;

  yolo_partial_kernel<<<blocks, threads, 0, stream>>>(pred, targ, partial, ngroups);
  yolo_final_kernel<<<1, threads, 0, stream>>>(partial, blocks, out);
}